// NNDModule_56521769616124
// MI455X (gfx1250) — compile-verified
//
#include <hip/hip_runtime.h>

typedef float v2f __attribute__((ext_vector_type(2)));
typedef float v8f __attribute__((ext_vector_type(8)));

// ---------------------------------------------------------------------------
// Pass 0: augment reference points into WMMA-ready B-form:
//   bf[j] = [ -2x, -2y, -2z, ||p||^2 ]   (one float4 per point)
// This is computed ONCE per point instead of once per (row-tile, point) pair,
// removing ~7 VALU per WMMA from the hot loop of the main kernel.
// ---------------------------------------------------------------------------
__global__ __launch_bounds__(256) void augment_kernel(
    const float* __restrict__ pts,   // [total, 3]
    float4* __restrict__ bform,      // [total] of float4
    int total)
{
    int i = blockIdx.x * 256 + threadIdx.x;
    if (i < total) {
        const float* p = pts + (size_t)i * 3;
        float x = p[0], y = p[1], z = p[2];
        float4 v;
        v.x = -2.0f * x;
        v.y = -2.0f * y;
        v.z = -2.0f * z;
        v.w = x * x + y * y + z * z;
        bform[i] = v;
    }
}

// ---------------------------------------------------------------------------
// Main kernel (precomputed-B variant). One wave handles 16 query points vs
// all M reference points using V_WMMA_F32_16X16X4_F32 with the augmented-K
// trick:  A row = [qx,qy,qz,1],  B col = [-2rx,-2ry,-2rz,||r||^2]
//   =>    D = -2 q.r + ||r||^2     (||q||^2 added after the column-min)
// Hot loop per 16-column chunk: 1x global_load_b64 + 1x WMMA + min3 tree.
// ---------------------------------------------------------------------------
__global__ __launch_bounds__(256) void chamfer_dir_pre_kernel(
    const float* __restrict__ q,      // [B, N, 3] raw query points
    const float* __restrict__ rbf,    // [B, M, 4] B-form reference points
    float* __restrict__ out,          // [B, N]
    int N, int M)
{
    const int lane = threadIdx.x & 31;
    const int waveInBlock = threadIdx.x >> 5;
    const int sub  = lane & 15;   // column / point index within half-wave
    const int half = lane >> 4;   // 0: K={0,1}, rows 0..7 ; 1: K={2,3}, rows 8..15

    const int tilesPerBatch = N >> 4;
    const int gtile = blockIdx.x * 8 + waveInBlock;
    const int b = gtile / tilesPerBatch;
    const int t = gtile - b * tilesPerBatch;

    // ---- A tile: load my query point once (built in registers, loop-invariant)
    const float* qp = q + ((size_t)b * N + (size_t)t * 16 + sub) * 3;
    const float qx = qp[0], qy = qp[1], qz = qp[2];
    const float nq = qx * qx + qy * qy + qz * qz;

    v2f A;
    A.x = half ? qz   : qx;   // K=2 : K=0
    A.y = half ? 1.0f : qy;   // K=3 : K=1

    v8f acc;
#pragma unroll
    for (int i = 0; i < 8; ++i) acc[i] = 3.0e38f;

    // B-form stream for this batch: lane loads the DWORD pair its K-slots need.
    const v2f* rb = (const v2f*)(rbf + (size_t)b * M * 4) + half; // pair select
#pragma unroll 4
    for (int c = 0; c < M; c += 16) {
        const v2f* rp = rb + (size_t)(c + sub) * 2;
        __builtin_prefetch((const float*)rp + 128, 0, 3);  // 16-pt chunk after next (32 pts x4 floats)
        v2f Bv = *rp;                                      // single b64 load

        v8f cz = {};
        v8f d = __builtin_amdgcn_wmma_f32_16x16x4_f32(
            false, A, false, Bv, (short)0, cz, false, false);
#pragma unroll
        for (int i = 0; i < 8; ++i) acc[i] = fminf(acc[i], d[i]);
    }

    // ---- Min across the 16 columns in each half (xor 1/2/4/8 stays in-half)
#pragma unroll
    for (int i = 0; i < 8; ++i) {
        float v = acc[i];
        v = fminf(v, __shfl_xor(v, 1, 32));
        v = fminf(v, __shfl_xor(v, 2, 32));
        v = fminf(v, __shfl_xor(v, 4, 32));
        v = fminf(v, __shfl_xor(v, 8, 32));
        acc[i] = v;
    }

    const float nqPartner = __shfl_xor(nq, 8, 32);
    const float myNq = half ? nqPartner : nq;

    float outVal = 0.0f;
    int   outRow = -1;
#pragma unroll
    for (int v = 0; v < 8; ++v) {
        if (sub == v) { outVal = acc[v]; outRow = v + 8 * half; }
    }
    if (outRow >= 0) {
        out[(size_t)b * N + (size_t)t * 16 + outRow] = fmaxf(0.0f, outVal + myNq);
    }
}

// ---------------------------------------------------------------------------
// Fallback (self-contained) variant: builds B in registers each chunk.
// Used only if the harness workspace is too small for the B-form arrays.
// ---------------------------------------------------------------------------
__global__ __launch_bounds__(256) void chamfer_dir_kernel(
    const float* __restrict__ q, const float* __restrict__ r,
    float* __restrict__ out, int N, int M)
{
    const int lane = threadIdx.x & 31;
    const int waveInBlock = threadIdx.x >> 5;
    const int sub  = lane & 15;
    const int half = lane >> 4;

    const int tilesPerBatch = N >> 4;
    const int gtile = blockIdx.x * 8 + waveInBlock;
    const int b = gtile / tilesPerBatch;
    const int t = gtile - b * tilesPerBatch;

    const float* qp = q + ((size_t)b * N + (size_t)t * 16 + sub) * 3;
    const float qx = qp[0], qy = qp[1], qz = qp[2];
    const float nq = qx * qx + qy * qy + qz * qz;

    v2f A;
    A.x = half ? qz   : qx;
    A.y = half ? 1.0f : qy;

    v8f acc;
#pragma unroll
    for (int i = 0; i < 8; ++i) acc[i] = 3.0e38f;

    const float* rbase = r + (size_t)b * M * 3;
    for (int c = 0; c < M; c += 16) {
        const float* rp = rbase + (size_t)(c + sub) * 3;
        __builtin_prefetch(rp + 48, 0, 3);
        const float rx = rp[0], ry = rp[1], rz = rp[2];
        const float nr = rx * rx + ry * ry + rz * rz;

        v2f Bv;
        Bv.x = half ? (-2.0f * rz) : (-2.0f * rx);
        Bv.y = half ?  nr          : (-2.0f * ry);

        v8f cz = {};
        v8f d = __builtin_amdgcn_wmma_f32_16x16x4_f32(
            false, A, false, Bv, (short)0, cz, false, false);
#pragma unroll
        for (int i = 0; i < 8; ++i) acc[i] = fminf(acc[i], d[i]);
    }

#pragma unroll
    for (int i = 0; i < 8; ++i) {
        float v = acc[i];
        v = fminf(v, __shfl_xor(v, 1, 32));
        v = fminf(v, __shfl_xor(v, 2, 32));
        v = fminf(v, __shfl_xor(v, 4, 32));
        v = fminf(v, __shfl_xor(v, 8, 32));
        acc[i] = v;
    }

    const float nqPartner = __shfl_xor(nq, 8, 32);
    const float myNq = half ? nqPartner : nq;

    float outVal = 0.0f;
    int   outRow = -1;
#pragma unroll
    for (int v = 0; v < 8; ++v) {
        if (sub == v) { outVal = acc[v]; outRow = v + 8 * half; }
    }
    if (outRow >= 0) {
        out[(size_t)b * N + (size_t)t * 16 + outRow] = fmaxf(0.0f, outVal + myNq);
    }
}

extern "C" void kernel_launch(void* const* d_in, const int* in_sizes, int n_in,
                              void* d_out, int out_size, void* d_ws, size_t ws_size,
                              hipStream_t stream) {
    (void)n_in; (void)out_size;
    const float* x1 = (const float*)d_in[0];   // [B, N, 3]
    const float* x2 = (const float*)d_in[1];   // [B, M, 3]
    float* out = (float*)d_out;                // dist1 [B,N] ++ dist2 [B,M]

    const int B = 16;
    const int N = in_sizes[0] / (B * 3);       // 4096
    const int M = in_sizes[1] / (B * 3);       // 4096

    dim3 block(256);                            // 8 waves/block (wave32)
    dim3 grid1((B * (N >> 4)) / 8);
    dim3 grid2((B * (M >> 4)) / 8);

    const size_t bytes1 = (size_t)B * N * 4 * sizeof(float);  // B-form of x1
    const size_t bytes2 = (size_t)B * M * 4 * sizeof(float);  // B-form of x2

    if (ws_size >= bytes1 + bytes2) {
        float4* bf1 = (float4*)d_ws;                          // [B*N]
        float4* bf2 = (float4*)((char*)d_ws + bytes1);        // [B*M]

        int tot1 = B * N, tot2 = B * M;
        augment_kernel<<<(tot1 + 255) / 256, block, 0, stream>>>(x1, bf1, tot1);
        augment_kernel<<<(tot2 + 255) / 256, block, 0, stream>>>(x2, bf2, tot2);

        // dist1: x1 queries vs x2 refs (precomputed B-form)
        chamfer_dir_pre_kernel<<<grid1, block, 0, stream>>>(
            x1, (const float*)bf2, out, N, M);
        // dist2: x2 queries vs x1 refs
        chamfer_dir_pre_kernel<<<grid2, block, 0, stream>>>(
            x2, (const float*)bf1, out + (size_t)B * N, M, N);
    } else {
        // Workspace too small: self-contained path
        chamfer_dir_kernel<<<grid1, block, 0, stream>>>(x1, x2, out, N, M);
        chamfer_dir_kernel<<<grid2, block, 0, stream>>>(x2, x1, out + (size_t)B * N, M, N);
    }
}